// CapsuleLayer_57011395887206
// MI455X (gfx1250) — compile-verified
//
#include <hip/hip_runtime.h>

typedef __attribute__((ext_vector_type(2))) float v2f;
typedef __attribute__((ext_vector_type(4))) float v4f;
typedef __attribute__((ext_vector_type(8))) float v8f;

#define N_CAPS1 1152
#define K_DIM   8
#define NJ      10
#define DD      16
#define OUT_PER_B (N_CAPS1 * NJ * DD)   // 184320 floats per batch element

// One wave32 computes a 16(batch) x 160(j*d) tile for one capsule n,
// as 10 chained-K (4+4) V_WMMA_F32_16X16X4_F32 tile products.
__global__ __launch_bounds__(256)
void caps_uhat_wmma(const float* __restrict__ x,   // [512, 1152*8]
                    const float* __restrict__ W,   // [1152, 10, 16, 8]
                    float* __restrict__ out)       // [512, 1152, 10, 16]
{
    const int lane  = threadIdx.x & 31;
    const int wib   = threadIdx.x >> 5;               // wave in block (0..7)
    const int w     = blockIdx.x * 8 + wib;           // global wave id
    const int bt    = w & 31;                         // batch tile (0..31), 16 rows each
    const int n     = w >> 5;                         // capsule (0..1151)
    const int half  = lane >> 4;                      // 0: K={0,1}/{4,5}+N 0..15 ; 1: K={2,3}/{6,7}
    const int m     = lane & 15;                      // row M (for A) / column N (for B,D)

    // ---- load this lane's pc row (b = bt*16 + m), 8 contiguous floats ----
    const int b = bt * 16 + m;
    const float* xr = x + (size_t)b * (N_CAPS1 * K_DIM) + (size_t)n * K_DIM;
    v4f x0 = *(const v4f*)(xr);       // k = 0..3
    v4f x1 = *(const v4f*)(xr + 4);   // k = 4..7

    // ---- squash: factor = |s|^2 / ((1+|s|^2) * sqrt(|s|^2 + eps)) ----
    float sq = x0.x*x0.x + x0.y*x0.y + x0.z*x0.z + x0.w*x0.w
             + x1.x*x1.x + x1.y*x1.y + x1.z*x1.z + x1.w*x1.w;
    float factor = sq / ((1.0f + sq) * sqrtf(sq + 1e-7f));

    // ---- A fragments per ISA 32-bit 16x4 layout (row m in this lane) ----
    v2f a_lo, a_hi;
    a_lo.x = (half ? x0.z : x0.x) * factor;   // K = 2 : 0
    a_lo.y = (half ? x0.w : x0.y) * factor;   // K = 3 : 1
    a_hi.x = (half ? x1.z : x1.x) * factor;   // K = 6 : 4
    a_hi.y = (half ? x1.w : x1.y) * factor;   // K = 7 : 5

    const float* wbase = W + (((size_t)n * NJ) * DD) * K_DIM;  // W[n,0,0,0]
    // D-tile store base: lane<16 -> rows 0..7 of tile, lane>=16 -> rows 8..15
    size_t obase = (((size_t)(bt * 16 + half * 8) * N_CAPS1 + n) * NJ) * DD + m;

    #pragma unroll
    for (int j = 0; j < NJ; ++j) {
        // ---- B fragment: column N=m is W[n,j,m,0..7]; pick K slice by half ----
        const float* wr = wbase + ((size_t)j * DD + m) * K_DIM;
        v2f b_lo = *(const v2f*)(wr + 2 * half);       // K = {0,1} or {2,3}
        v2f b_hi = *(const v2f*)(wr + 4 + 2 * half);   // K = {4,5} or {6,7}

        v8f c = {};
        c = __builtin_amdgcn_wmma_f32_16x16x4_f32(false, a_lo, false, b_lo,
                                                  (short)0, c, false, false);
        c = __builtin_amdgcn_wmma_f32_16x16x4_f32(false, a_hi, false, b_hi,
                                                  (short)0, c, false, false);

        // ---- stream out 16x16 tile; write-once data -> non-temporal ----
        size_t o = obase + (size_t)j * DD;
        #pragma unroll
        for (int v = 0; v < 8; ++v)
            __builtin_nontemporal_store(c[v], out + o + (size_t)v * OUT_PER_B);
    }
}

extern "C" void kernel_launch(void* const* d_in, const int* in_sizes, int n_in,
                              void* d_out, int out_size, void* d_ws, size_t ws_size,
                              hipStream_t stream) {
    const float* x = (const float*)d_in[0];   // [512, 9216] fp32
    const float* W = (const float*)d_in[1];   // [1, 1152, 10, 16, 8] fp32
    float* out = (float*)d_out;               // [512, 1152, 10, 16, 1] fp32

    // 1152 capsules * 32 batch-tiles = 36864 waves = 4608 blocks of 8 waves
    dim3 grid(4608), block(256);
    caps_uhat_wmma<<<grid, block, 0, stream>>>(x, W, out);
}